// QuadPool_16458314678304
// MI455X (gfx1250) — compile-verified
//
#include <hip/hip_runtime.h>

// QuadPool: masked 4-child gather-average, NP=1e6 parents, C=64 channels.
// Bandwidth-bound: ~16MB idx + ~1.02GB gathered rows + 256MB out ~= 1.3GB
// => ~56us floor @ 23.3 TB/s. CDNA5 path: GLOBAL_LOAD_ASYNC_TO_LDS_B128
// per-lane gathers (ASYNCcnt) staged into LDS, cheap VALU masked reduce,
// non-temporal output stores to preserve L2 for the random gathers.
// No GEMM structure (per-parent B operand) -> WMMA intentionally not used.

typedef float v4f __attribute__((ext_vector_type(4)));  // native vector: OK for NT builtins
typedef int   v4i __attribute__((ext_vector_type(4)));

constexpr int kC         = 64;   // channels
constexpr int kParPerBlk = 16;   // parents per block
constexpr int kThreads   = 256;  // 8 waves (wave32)

__global__ __launch_bounds__(kThreads)
void quadpool_async_kernel(const float* __restrict__ feats,
                           const int*   __restrict__ cidx,
                           float*       __restrict__ out,
                           int np)
{
    // [parent][child][chunk] : 16 * 4 * 16 v4f = 16 KB
    __shared__ v4f tile[kParPerBlk * 4 * (kC / 4)];

    const int t  = threadIdx.x;
    const int p0 = blockIdx.x * kParPerBlk;

    // LDS allocation-relative byte offset of the tile (low 32 bits of the
    // generic pointer == addrspace(3) offset; HW adds the wave LDS base).
    const unsigned lds_base = (unsigned)(unsigned long long)(uintptr_t)&tile[0];

    // ---- stage A: batch the 4 index loads (one clause, one wait) ----
    int idxreg[4];
#pragma unroll
    for (int i = 0; i < 4; ++i) {
        const int linear = t + i * kThreads;   // float4 slot 0..1023
        const int pl     = linear >> 6;        // local parent  0..15
        const int child  = (linear >> 4) & 3;  // child slot    0..3
        int gp = p0 + pl;
        if (gp >= np) gp = np - 1;             // tail clamp (NP%16==0 anyway)
        idxreg[i] = cidx[(size_t)gp * 4 + child];
    }

    // ---- stage B: issue 4 async B128 gathers back-to-back ----
    // 1024 lane-transfers per block; 16 consecutive lanes cover one full
    // 256B child row (coalesced). -1 sentinel -> row 0, masked in consume.
#pragma unroll
    for (int i = 0; i < 4; ++i) {
        const int linear = t + i * kThreads;
        const int chunk  = linear & 15;        // float4 chunk within row
        const int safe   = idxreg[i] < 0 ? 0 : idxreg[i];

        const unsigned long long gaddr =
            (unsigned long long)(uintptr_t)(feats + (size_t)safe * kC + chunk * 4);
        const unsigned laddr = lds_base + (unsigned)(linear * 16);

        // GLOBAL_LOAD_ASYNC_TO_LDS_B128: VDST = per-lane LDS byte addr,
        // VADDR = 64-bit global addr (GV mode). Tracked by ASYNCcnt.
        asm volatile("global_load_async_to_lds_b128 %0, %1, off"
                     :: "v"(laddr), "v"(gaddr)
                     : "memory");
    }

#if __has_builtin(__builtin_amdgcn_s_wait_asynccnt)
    __builtin_amdgcn_s_wait_asynccnt(0);
#else
    asm volatile("s_wait_asynccnt 0" ::: "memory");
#endif
    __syncthreads();   // LDS written by all 8 waves' async ops

    // ---- consume: 16 lanes per parent, one float4 chunk per lane ----
    const int pl    = t >> 4;
    const int chunk = t & 15;
    const int gp    = p0 + pl;
    if (gp < np) {
        // one b128 broadcast load of this parent's 4 indices (L1-hot)
        const v4i pi = reinterpret_cast<const v4i*>(cidx)[gp];

        v4f   acc = (v4f)(0.0f);
        float cnt = 0.f;
#pragma unroll
        for (int j = 0; j < 4; ++j) {
            const float m = pi[j] >= 0 ? 1.0f : 0.0f;
            const v4f   v = tile[pl * 64 + j * 16 + chunk]; // 16B stride: conflict-free
            acc += m * v;
            cnt += m;
        }
        acc *= 1.0f / fmaxf(cnt, 1.0f);

        v4f* dst = reinterpret_cast<v4f*>(out) + (size_t)gp * (kC / 4) + chunk;
#if __has_builtin(__builtin_nontemporal_store)
        // Write-once output: NT store keeps the 192MB L2 for gather reuse.
        __builtin_nontemporal_store(acc, dst);
#else
        *dst = acc;
#endif
    }
}

extern "C" void kernel_launch(void* const* d_in, const int* in_sizes, int n_in,
                              void* d_out, int out_size, void* d_ws, size_t ws_size,
                              hipStream_t stream)
{
    const float* feats = (const float*)d_in[0];   // child_features (NC, 64) f32
    const int*   cidx  = (const int*)d_in[1];     // children_idx  (NP, 4)  int
    float*       out   = (float*)d_out;           // (NP, 64) f32

    const int np     = in_sizes[1] / 4;
    const int blocks = (np + kParPerBlk - 1) / kParPerBlk;
    quadpool_async_kernel<<<blocks, kThreads, 0, stream>>>(feats, cidx, out, np);
}